// PosedPointmapChamferLoss_51427938402462
// MI455X (gfx1250) — compile-verified
//
#include <hip/hip_runtime.h>
#include <math.h>

typedef float v2f __attribute__((ext_vector_type(2)));
typedef float v8f __attribute__((ext_vector_type(8)));

#define BATCH 2
#define NV1 8192            // gt points  (columns / N)
#define NV2 8192            // pred points (rows / M)
#define ROWS_PER_BLOCK 128  // 8 waves * 16 rows
#define COLSPLIT 4
#define COLS_PER_BLOCK (NV1 / COLSPLIT) // 2048
#define CHUNK 512           // gt points staged in LDS per iteration (8 KB)
#define INF_F 3.0e38f

// -------- prep: pack gt as (-2x,-2y,-2z,|q|^2), pred norms, init mins --------
__global__ void chamfer_prep(const float* __restrict__ v_gt,
                             const float* __restrict__ v_pred,
                             float4* __restrict__ gtpk,
                             float* __restrict__ prednorm,
                             float* __restrict__ colmin,
                             float* __restrict__ rowmin) {
    int i = blockIdx.x * blockDim.x + threadIdx.x;
    if (i >= BATCH * NV1) return;
    const float* g = v_gt + (size_t)i * 3;
    float gx = g[0], gy = g[1], gz = g[2];
    gtpk[i] = make_float4(-2.0f * gx, -2.0f * gy, -2.0f * gz,
                          gx * gx + gy * gy + gz * gz);
    const float* p = v_pred + (size_t)i * 3;
    float px = p[0], py = p[1], pz = p[2];
    prednorm[i] = px * px + py * py + pz * pz;
    colmin[i] = INF_F;
    rowmin[i] = INF_F;
}

// -------- main tile kernel: WMMA d^2 tiles + fused min reductions --------
__global__ __launch_bounds__(256)
void chamfer_tiles(const float* __restrict__ v_pred,
                   const float4* __restrict__ gtpk,
                   const float* __restrict__ prednorm,
                   float* __restrict__ colmin,
                   float* __restrict__ rowmin) {
    const int blk    = blockIdx.x;
    const int cs     = blk % COLSPLIT;
    const int t      = blk / COLSPLIT;
    const int rowblk = t % (NV2 / ROWS_PER_BLOCK);
    const int b      = t / (NV2 / ROWS_PER_BLOCK);
    const int rowbase = rowblk * ROWS_PER_BLOCK;
    const int colbase = cs * COLS_PER_BLOCK;

    const int lane = threadIdx.x & 31;
    const int wave = threadIdx.x >> 5;
    const int half = lane >> 4;   // 0: lanes 0-15, 1: lanes 16-31
    const int l16  = lane & 15;
    const int wrow = rowbase + wave * 16;  // this wave's 16 pred rows

    // A matrix 16x4 f32 layout: vgpr0 = K0 (lanes 0-15) / K2 (lanes 16-31),
    //                           vgpr1 = K1 / K3.  Row M = l16 on both halves.
    // Homogeneous encoding: row = (px, py, pz, 1).
    v2f Amat;
    {
        const int arow = wrow + l16;
        const float* pp = v_pred + ((size_t)b * NV2 + arow) * 3;
        if (half == 0) { Amat.x = pp[0]; Amat.y = pp[1]; }
        else           { Amat.x = pp[2]; Amat.y = 1.0f;  }
    }

    // C init = |p_M|^2 : C layout vgpr r -> M = r + 8*half, N = l16
    v8f Cinit;
    {
        const float* pn = prednorm + (size_t)b * NV2 + wrow + half * 8;
        #pragma unroll
        for (int r = 0; r < 8; ++r) Cinit[r] = pn[r];
    }

    v8f rm;  // per-lane running row-mins over this block's columns
    #pragma unroll
    for (int r = 0; r < 8; ++r) rm[r] = INF_F;

    __shared__ float4 sgt[CHUNK];           // 8 KB gt staging
    __shared__ float  scol[COLS_PER_BLOCK]; // 8 KB per-block column mins

    // init per-block LDS column mins
    for (int tt = threadIdx.x; tt < COLS_PER_BLOCK; tt += 256)
        scol[tt] = INF_F;

    for (int chunk = 0; chunk < COLS_PER_BLOCK; chunk += CHUNK) {
        for (int tt = threadIdx.x; tt < CHUNK; tt += 256)
            sgt[tt] = gtpk[(size_t)b * NV1 + colbase + chunk + tt];
        __syncthreads();

        // B matrix 4x16 f32 layout: vgpr0 = K0 row (lanes 0-15) / K2 row,
        // vgpr1 = K1 / K3.  Col N = l16. Packed gt: (-2x,-2y,-2z,|q|^2).
        // Software-pipelined: load tile jt+16 while WMMA'ing tile jt.
        const int boff = half * 2;
        v2f Bcur;
        {
            const float* g = (const float*)&sgt[l16];
            Bcur.x = g[boff]; Bcur.y = g[boff + 1];
        }
        for (int jt = 0; jt < CHUNK; jt += 16) {
            const int nj = (jt + 16) & (CHUNK - 1);   // wraps to 0 on last iter
            v2f Bnext;
            {
                const float* g = (const float*)&sgt[nj + l16];
                Bnext.x = g[boff]; Bnext.y = g[boff + 1];
            }

            v8f D = __builtin_amdgcn_wmma_f32_16x16x4_f32(
                false, Amat, false, Bcur, (short)0, Cinit, false, false);

            // row-min: defer cross-lane part, accumulate per-lane column subset
            #pragma unroll
            for (int r = 0; r < 8; ++r) rm[r] = __builtin_fminf(rm[r], D[r]);

            // col-min over the 8 rows held by this lane; both halves hit the
            // same LDS word (ds_min_num_f32, no return, no divergence)
            float cm = __builtin_fminf(D[0], D[1]);
            cm = __builtin_fminf(cm, __builtin_fminf(D[2], D[3]));
            cm = __builtin_fminf(cm, __builtin_fminf(D[4], D[5]));
            cm = __builtin_fminf(cm, __builtin_fminf(D[6], D[7]));
            atomicMin(&scol[chunk + jt + l16], cm);

            Bcur = Bnext;
        }
        __syncthreads();
    }

    // flush per-block column mins to global (1/8 the atomics of per-tile)
    for (int tt = threadIdx.x; tt < COLS_PER_BLOCK; tt += 256)
        atomicMin(&colmin[(size_t)b * NV1 + colbase + tt], scol[tt]);

    // finalize row mins: butterfly across each 16-lane half, merge col-splits
    #pragma unroll
    for (int r = 0; r < 8; ++r) {
        float v = rm[r];
        v = __builtin_fminf(v, __shfl_xor(v, 1, 32));
        v = __builtin_fminf(v, __shfl_xor(v, 2, 32));
        v = __builtin_fminf(v, __shfl_xor(v, 4, 32));
        v = __builtin_fminf(v, __shfl_xor(v, 8, 32));
        if (l16 == 0)
            atomicMin(&rowmin[(size_t)b * NV2 + wrow + r + half * 8], v);
    }
}

// -------- final deterministic reduction to the two scalar outputs --------
__global__ __launch_bounds__(256)
void chamfer_finalize(const float* __restrict__ rowmin,
                      const float* __restrict__ colmin,
                      const float* __restrict__ mask,
                      float* __restrict__ out) {
    __shared__ float s1[256];
    __shared__ float s2[256];
    float a1 = 0.0f, a2 = 0.0f;
    for (int i = threadIdx.x; i < BATCH * NV2; i += 256) a1 += rowmin[i] * mask[i];
    for (int i = threadIdx.x; i < BATCH * NV1; i += 256) a2 += colmin[i];
    s1[threadIdx.x] = a1;
    s2[threadIdx.x] = a2;
    __syncthreads();
    for (int s = 128; s > 0; s >>= 1) {
        if ((int)threadIdx.x < s) {
            s1[threadIdx.x] += s1[threadIdx.x + s];
            s2[threadIdx.x] += s2[threadIdx.x + s];
        }
        __syncthreads();
    }
    if (threadIdx.x == 0) {
        float m1 = s1[0] / (float)(BATCH * NV2);  // masked pred->gt mean
        float m2 = s2[0] / (float)(BATCH * NV1);  // gt->pred mean
        out[0] = m1 + m2;  // loss
        out[1] = m1;       // masked_loss_v_pred_to_v
    }
}

extern "C" void kernel_launch(void* const* d_in, const int* in_sizes, int n_in,
                              void* d_out, int out_size, void* d_ws, size_t ws_size,
                              hipStream_t stream) {
    const float* v_gt   = (const float*)d_in[0];  // (B, V1, 3)
    const float* v_pred = (const float*)d_in[1];  // (B, V2, 3)
    const float* mask   = (const float*)d_in[2];  // (B, V2)
    float* out = (float*)d_out;

    char* ws = (char*)d_ws;
    float4* gtpk     = (float4*)ws;                                   // 256 KB
    float*  prednorm = (float*)(ws + (size_t)BATCH * NV1 * 16);       //  64 KB
    float*  colmin   = (float*)(ws + (size_t)BATCH * NV1 * 16
                                   + (size_t)BATCH * NV2 * 4);        //  64 KB
    float*  rowmin   = (float*)(ws + (size_t)BATCH * NV1 * 16
                                   + (size_t)BATCH * NV2 * 4
                                   + (size_t)BATCH * NV1 * 4);        //  64 KB

    chamfer_prep<<<(BATCH * NV1 + 255) / 256, 256, 0, stream>>>(
        v_gt, v_pred, gtpk, prednorm, colmin, rowmin);

    chamfer_tiles<<<BATCH * (NV2 / ROWS_PER_BLOCK) * COLSPLIT, 256, 0, stream>>>(
        v_pred, gtpk, prednorm, colmin, rowmin);

    chamfer_finalize<<<1, 256, 0, stream>>>(rowmin, colmin, mask, out);
}